// DenseGrid_32177894982357
// MI455X (gfx1250) — compile-verified
//
#include <hip/hip_runtime.h>
#include <math.h>

typedef float v4f __attribute__((ext_vector_type(4)));

#define NLOD   8
#define NSTAGE 4   // LODs 0..3 staged in LDS (~37.7 KB, < 64 KB dynamic-LDS default)

__global__ __launch_bounds__(256) void dense_grid_gather_kernel(
    const float* __restrict__ pts,
    const float* __restrict__ c0, const float* __restrict__ c1,
    const float* __restrict__ c2, const float* __restrict__ c3,
    const float* __restrict__ c4, const float* __restrict__ c5,
    const float* __restrict__ c6, const float* __restrict__ c7,
    float* __restrict__ out, int n,
    int r0, int r1, int r2, int r3, int r4, int r5, int r6, int r7)
{
    extern __shared__ float smem[];

    const float* cbs[NLOD] = {c0, c1, c2, c3, c4, c5, c6, c7};
    const int    res[NLOD] = {r0, r1, r2, r3, r4, r5, r6, r7};

    // LDS float-offsets of each staged codebook
    int off[NSTAGE];
    {
        int o = 0;
#pragma unroll
        for (int L = 0; L < NSTAGE; ++L) { off[L] = o; o += res[L] * res[L] * 2; }
    }

    // ---- Stage small codebooks into LDS with CDNA5 async copies ------------
    // global_load_async_to_lds_b64: per-lane 8-byte Memory->LDS DMA, tracked
    // by ASYNCcnt (no VGPR data path).
    const int tid = threadIdx.x;
    const unsigned lds_base = (unsigned)(unsigned long long)(uintptr_t)smem;
#pragma unroll
    for (int L = 0; L < NSTAGE; ++L) {
        const int      chunks = res[L] * res[L];      // one float2 (8B) per entry
        const float*   g      = cbs[L];
        const unsigned base   = lds_base + (unsigned)off[L] * 4u;
        for (int c = tid; c < chunks; c += (int)blockDim.x) {
            const unsigned laddr = base + (unsigned)c * 8u;
            const float*   gp    = g + (size_t)c * 2;
            asm volatile("global_load_async_to_lds_b64 %0, %1, off"
                         :: "v"(laddr), "v"(gp) : "memory");
        }
    }
    asm volatile("s_wait_asynccnt 0" ::: "memory");   // my async copies landed
    __syncthreads();                                  // everyone's copies landed

    // ---- Gather phase ------------------------------------------------------
    const float2* __restrict__ pts2 = (const float2*)pts;
    const int stride = (int)(gridDim.x * blockDim.x);

    for (int i = (int)(blockIdx.x * blockDim.x) + tid; i < n; i += stride) {
        const float2 p = pts2[i];                     // coalesced b64 load
        float fa[NLOD], fb[NLOD];
#pragma unroll
        for (int L = 0; L < NLOD; ++L) {
            const int r   = res[L];
            const int xi  = __float2int_rd(p.x * (float)(r - 1)); // floor, x in [0,1)
            const int yi  = __float2int_rd(p.y * (float)(r - 1));
            const int idx = xi + yi * r;
            float2 v;
            if (L < NSTAGE) {
                v = *(const float2*)(&smem[off[L] + 2 * idx]);    // ds_load_b64
            } else {
                v = ((const float2*)cbs[L])[idx];                 // L2-resident gather
            }
            fa[L] = v.x;
            fb[L] = v.y;
        }
        // out[n, f*8 + L]: positions 0..7 = feat0 of L0..L7, 8..15 = feat1
        v4f o0 = {fa[0], fa[1], fa[2], fa[3]};
        v4f o1 = {fa[4], fa[5], fa[6], fa[7]};
        v4f o2 = {fb[0], fb[1], fb[2], fb[3]};
        v4f o3 = {fb[4], fb[5], fb[6], fb[7]};
        v4f* ov = (v4f*)(out + (size_t)i * 16);       // 64B-aligned
        __builtin_nontemporal_store(o0, ov + 0);      // streaming: keep L2 for codebooks
        __builtin_nontemporal_store(o1, ov + 1);
        __builtin_nontemporal_store(o2, ov + 2);
        __builtin_nontemporal_store(o3, ov + 3);
    }
}

extern "C" void kernel_launch(void* const* d_in, const int* in_sizes, int n_in,
                              void* d_out, int out_size, void* d_ws, size_t ws_size,
                              hipStream_t stream) {
    (void)n_in; (void)out_size; (void)d_ws; (void)ws_size;

    const float* pts = (const float*)d_in[0];
    const float* cb[NLOD];
    int res[NLOD];
    for (int i = 0; i < NLOD; ++i) {
        cb[i] = (const float*)d_in[1 + i];
        const int entries = in_sizes[1 + i] / 2;              // feat_dim = 2
        res[i] = (int)(sqrt((double)entries) + 0.5);          // exact: entries = res^2
    }
    const int n = in_sizes[0] / 2;                            // pts is [N,2]
    if (n <= 0) return;

    size_t smem_bytes = 0;
    for (int L = 0; L < NSTAGE; ++L)
        smem_bytes += (size_t)res[L] * res[L] * 2 * sizeof(float);

    const int threads = 256;   // 8 wave32s per block
    const int blocks  = 2048;  // grid-stride: ~7.6 points/thread, bounds LDS restaging

    dense_grid_gather_kernel<<<blocks, threads, smem_bytes, stream>>>(
        pts, cb[0], cb[1], cb[2], cb[3], cb[4], cb[5], cb[6], cb[7],
        (float*)d_out, n,
        res[0], res[1], res[2], res[3], res[4], res[5], res[6], res[7]);
}